// CostumeLIF_6244882448890
// MI455X (gfx1250) — compile-verified
//
#include <hip/hip_runtime.h>

#define T_STEPS 32
#define TPB 256
#define PIPE 8   // async pipeline depth: 8 outstanding 512B loads per wave

typedef float v4f __attribute__((ext_vector_type(4)));
typedef int   v4i __attribute__((vector_size(4 * sizeof(int))));

#if defined(__HIP_DEVICE_COMPILE__) && defined(__gfx1250__) && \
    __has_builtin(__builtin_amdgcn_global_load_async_to_lds_b128)
#define USE_ASYNC_LDS 1
typedef __attribute__((address_space(1))) v4i* gv4_p;  // global
typedef __attribute__((address_space(3))) v4i* lv4_p;  // LDS

// s_wait_asynccnt needs a literal imm; 'n' is compile-time constant after the
// full unroll, so this switch folds to a single SOPP.
__device__ __forceinline__ void wait_async_le(int n) {
    switch (n) {
        case 0: asm volatile("s_wait_asynccnt 0" ::: "memory"); break;
        case 1: asm volatile("s_wait_asynccnt 1" ::: "memory"); break;
        case 2: asm volatile("s_wait_asynccnt 2" ::: "memory"); break;
        case 3: asm volatile("s_wait_asynccnt 3" ::: "memory"); break;
        case 4: asm volatile("s_wait_asynccnt 4" ::: "memory"); break;
        case 5: asm volatile("s_wait_asynccnt 5" ::: "memory"); break;
        case 6: asm volatile("s_wait_asynccnt 6" ::: "memory"); break;
        default: asm volatile("s_wait_asynccnt 7" ::: "memory"); break;
    }
}
#else
#define USE_ASYNC_LDS 0
#endif

// LIF scan: v += (x_t - v)/tau; s = H(v - 1); v *= (1 - s).
// Pure streaming problem (134 MB @ 23.3 TB/s ~ 5.8us floor, ~67 MFLOP total).
// Deep (8-stage) CDNA5 async global->LDS DMA pipeline keeps ~8 x 512B per wave
// in flight so per-WGP outstanding bytes reach the latency-bandwidth product;
// neuron state 'v' lives in VGPRs for all 32 steps; no barriers needed (each
// lane consumes exactly the 16B it DMA'd, ordered by ASYNCcnt/DScnt alone).
__global__ __launch_bounds__(TPB) void lif_scan_kernel(const float* __restrict__ x,
                                                       float* __restrict__ out,
                                                       int bn) {
    const int tid  = threadIdx.x;
    const int base = (blockIdx.x * TPB + tid) * 4;
    if (base + 4 > bn) return;

    const float* gx = x + base;
    float*       gy = out + base;

    v4f v = {0.0f, 0.0f, 0.0f, 0.0f};

#if USE_ASYNC_LDS
    __shared__ v4f buf[PIPE][TPB];   // 8 * 4KB = 32KB of the 320KB WGP pool
    // Prologue: stage t = 0..PIPE-1  (ASYNCcnt ramps to PIPE).
#pragma unroll
    for (int d = 0; d < PIPE; ++d)
        __builtin_amdgcn_global_load_async_to_lds_b128(
            (gv4_p)(gx + (size_t)d * bn), (lv4_p)&buf[d][tid], 0, 0);
#endif

#pragma unroll
    for (int t = 0; t < T_STEPS; ++t) {
#if USE_ASYNC_LDS
        // Async loads complete in order; after consuming loads 0..t-1, loads
        // t..min(t+PIPE-1, T-1) are outstanding. Waiting down to
        // min(PIPE-1, T-1-t) guarantees load 't' has landed.
        {
            const int remain = T_STEPS - 1 - t;
            wait_async_le(remain < PIPE - 1 ? remain : PIPE - 1);
        }
        v4f xt = buf[t % PIPE][tid];                 // ds_load_b128
        asm volatile("s_wait_dscnt 0" ::: "memory"); // data in VGPRs: slot reusable
        if (t + PIPE < T_STEPS)
            __builtin_amdgcn_global_load_async_to_lds_b128(
                (gv4_p)(gx + (size_t)(t + PIPE) * bn),
                (lv4_p)&buf[t % PIPE][tid], 0, 0);
#else
        v4f xt = *(const v4f*)(gx + (size_t)t * bn);
#endif
        v4f s;
        {
            float nv = v.x + (xt.x - v.x) * 0.5f;
            s.x = (nv - 1.0f) > 0.0f ? 1.0f : 0.0f;
            v.x = (s.x == 0.0f) ? nv : 0.0f;
        }
        {
            float nv = v.y + (xt.y - v.y) * 0.5f;
            s.y = (nv - 1.0f) > 0.0f ? 1.0f : 0.0f;
            v.y = (s.y == 0.0f) ? nv : 0.0f;
        }
        {
            float nv = v.z + (xt.z - v.z) * 0.5f;
            s.z = (nv - 1.0f) > 0.0f ? 1.0f : 0.0f;
            v.z = (s.z == 0.0f) ? nv : 0.0f;
        }
        {
            float nv = v.w + (xt.w - v.w) * 0.5f;
            s.w = (nv - 1.0f) > 0.0f ? 1.0f : 0.0f;
            v.w = (s.w == 0.0f) ? nv : 0.0f;
        }
        // Write-once output: non-temporal so 64MB of spikes doesn't evict the
        // input stream from L2.
        __builtin_nontemporal_store(s, (v4f*)(gy + (size_t)t * bn));
    }
}

extern "C" void kernel_launch(void* const* d_in, const int* in_sizes, int n_in,
                              void* d_out, int out_size, void* d_ws, size_t ws_size,
                              hipStream_t stream) {
    const float* x   = (const float*)d_in[0];
    float*       out = (float*)d_out;

    const int total = in_sizes[0];          // T * B * N
    const int bn    = total / T_STEPS;      // B * N elements per time step

    const int threads = (bn + 3) / 4;       // one thread per float4
    const int blocks  = (threads + TPB - 1) / TPB;

    lif_scan_kernel<<<blocks, TPB, 0, stream>>>(x, out, bn);
}